// MaxUnpooling2D_44976897524152
// MI455X (gfx1250) — compile-verified
//
#include <hip/hip_runtime.h>

// MaxUnpooling2D scatter-add for fixed shapes:
// updates/mask: [B=16, H=128, W=128, C=64]  (flat 16,777,216)
// output:       [B=16, Ho=256, Wo=256, C=64] (flat 67,108,864)
// All strides are powers of two -> pure shift/mask index decode.

typedef float v4f __attribute__((ext_vector_type(4)));
typedef int   v4i __attribute__((ext_vector_type(4)));

namespace {
constexpr int kHoM1 = 255;
constexpr int kWoM1 = 255;
}

// ---------------------------------------------------------------------------
// Kernel 1: zero the output with wide b128 stores (default RT temporal hint so
// the zeroed lines stay resident in the 192MB L2 for the atomic phase).
// ---------------------------------------------------------------------------
__global__ __launch_bounds__(256) void unpool_zero_kernel(v4f* __restrict__ out,
                                                          int n4) {
    int i = blockIdx.x * blockDim.x + threadIdx.x;
    if (i < n4) {
        v4f z = {0.0f, 0.0f, 0.0f, 0.0f};
        out[i] = z;
    }
}

// ---------------------------------------------------------------------------
// Kernel 2: scatter-add. One thread handles 4 consecutive channel elements:
//   - 1x non-temporal global_load_b128 of updates
//   - 1x non-temporal global_load_b128 of mask indices
//   - 4x no-return global_atomic_add_f32 into the output
// Non-temporal loads keep the streamed-once inputs from evicting the zeroed
// output lines out of L2.
// ---------------------------------------------------------------------------
__global__ __launch_bounds__(256) void unpool_scatter_kernel(
    const v4f* __restrict__ upd,
    const v4i* __restrict__ mask,
    float*     __restrict__ out,
    int n4) {
    int i = blockIdx.x * blockDim.x + threadIdx.x;
    if (i >= n4) return;

    v4f u = __builtin_nontemporal_load(&upd[i]);
    v4i m = __builtin_nontemporal_load(&mask[i]);

    // Flat input element index of lane 0 of this vec4.
    int e = i << 2;
    int b = e >> 20;        // e / (H*W*C) = e / 2^20
    int f = e & 63;         // channel of lane 0; lanes are f, f+1, f+2, f+3
    int base = (b << 22);   // b * Ho*Wo*C = b * 2^22

#pragma unroll
    for (int j = 0; j < 4; ++j) {
        int mm = m[j];
        int y = mm >> 14;           // m / (Wo*C)
        int x = (mm >> 6) & kWoM1;  // (m / C) % Wo
        y = min(max(y, 0), kHoM1);  // match reference clip
        // x already in [0,255] via mask, clamp is identity.
        int idx = base + (y << 14) + (x << 6) + (f + j);
        // Relaxed, device-scope, no-return float atomic add; steer clang to
        // the native global_atomic_add_f32 (no CAS loop).
        [[clang::atomic(no_remote_memory, no_fine_grained_memory,
                        ignore_denormal_mode)]] {
            (void)__hip_atomic_fetch_add(&out[idx], u[j], __ATOMIC_RELAXED,
                                         __HIP_MEMORY_SCOPE_AGENT);
        }
    }
}

extern "C" void kernel_launch(void* const* d_in, const int* in_sizes, int n_in,
                              void* d_out, int out_size, void* d_ws, size_t ws_size,
                              hipStream_t stream) {
    const float* upd  = (const float*)d_in[0];
    const int*   mask = (const int*)d_in[1];
    float*       out  = (float*)d_out;

    const int n_in_elems  = in_sizes[0];           // 16,777,216
    const int n_out_elems = out_size;              // 67,108,864

    // Phase 1: zero the output (atomics accumulate, so this must run per call).
    {
        int n4 = n_out_elems >> 2;                 // 16,777,216 float4 stores
        int blocks = (n4 + 255) / 256;
        unpool_zero_kernel<<<blocks, 256, 0, stream>>>((v4f*)out, n4);
    }

    // Phase 2: scatter-add (stream-ordered after the zero fill).
    {
        int n4 = n_in_elems >> 2;                  // 4,194,304 threads
        int blocks = (n4 + 255) / 256;
        unpool_scatter_kernel<<<blocks, 256, 0, stream>>>(
            (const v4f*)upd, (const v4i*)mask, out, n4);
    }
}